// LinearNoGate_24189255811502
// MI455X (gfx1250) — compile-verified
//
#include <hip/hip_runtime.h>

// LinearNoGate (e3nn-style per-irrep Linear) for irreps [(256,0),(128,1),(64,2)]
// out_block = (x_block @ W) / sqrt(mul); bias added to the leading 0e block.
// Three strided GEMMs on V_WMMA_F32_16X16X4_F32 (full fp32).
//   - W column-chunk staged in LDS (pair-major) -> B fragments are ds_load_b64
//   - 4 column tiles accumulated per K step -> 1 A load feeds 4 WMMAs

typedef float v2f __attribute__((ext_vector_type(2)));
typedef float v8f __attribute__((ext_vector_type(8)));

#define ROW_STRIDE 960
#define NNODES     100000

template <int MUL, int D, bool ADD_BIAS>
__global__ __launch_bounds__(256)
void irrep_linear_wmma(const float* __restrict__ xin,
                       const float* __restrict__ W,
                       const float* __restrict__ bias,
                       float* __restrict__ out,
                       int blockOff, float scale)
{
    constexpr int R           = NNODES * D;      // virtual rows (node, component)
    constexpr int ROW_TILES   = R / 16;          // exact: 6250 / 18750 / 31250
    constexpr int CHUNK_COLS  = 64;              // 4 x 16-wide column tiles
    constexpr int CHUNK_TILES = 4;
    constexpr int N_CHUNKS    = MUL / CHUNK_COLS;

    // Pair-major W slice: ldsW[(u/2)*CHUNK_COLS + c][2] = { W[u][c], W[u+1][c] }
    extern __shared__ float ldsW[];              // MUL * CHUNK_COLS floats

    const int tid  = threadIdx.x;
    const int lane = tid & 31;                   // wave32
    const int wave = tid >> 5;
    int rowTile = blockIdx.x * 8 + wave;
    const bool active = rowTile < ROW_TILES;     // wave-uniform
    if (!active) rowTile = ROW_TILES - 1;        // redundant compute, stores masked

    // --- A fragment mapping (16x4 f32, ISA 7.12.2) ---
    // lanes 0-15 : M = lane,    VGPR0 = K0, VGPR1 = K1
    // lanes 16-31: M = lane-16, VGPR0 = K2, VGPR1 = K3
    const int m    = lane & 15;
    const int half = lane >> 4;                  // 0 or 1 (K offset = 2*half)
    const int rA    = rowTile * 16 + m;
    const int nodeA = rA / D;
    const int ddA   = rA - nodeA * D;
    const float* aBase = xin + (size_t)nodeA * ROW_STRIDE + blockOff + ddA; // + u*D

    // D/C mapping: VGPR v -> M = v + 8*half, N = lane&15
    const int rDBase = rowTile * 16 + half * 8;

    const v2f* ldsB = (const v2f*)ldsW;

    for (int chunk = 0; chunk < N_CHUNKS; ++chunk) {
        const int colBase = chunk * CHUNK_COLS;

        __syncthreads();                         // previous chunk fully consumed
        // Cooperative fill: coalesced global reads, conflict-free strided LDS writes.
        #pragma unroll 4
        for (int i = tid; i < MUL * CHUNK_COLS; i += 256) {
            const int u = i >> 6;                // W row   (CHUNK_COLS == 64)
            const int c = i & (CHUNK_COLS - 1);  // W column within chunk
            ldsW[((u >> 1) * CHUNK_COLS + c) * 2 + (u & 1)] =
                W[(size_t)u * MUL + colBase + c];
        }
        __syncthreads();

        v8f acc[CHUNK_TILES];
        #pragma unroll
        for (int t = 0; t < CHUNK_TILES; ++t)
            acc[t] = v8f{0.f, 0.f, 0.f, 0.f, 0.f, 0.f, 0.f, 0.f};

        #pragma unroll 2
        for (int k0 = 0; k0 < MUL; k0 += 4) {
            const int kk = k0 + half * 2;        // even
            v2f a;
            a.x = aBase[kk * D];
            a.y = aBase[(kk + 1) * D];
            const int pBase = (kk >> 1) * CHUNK_COLS + m;
            #pragma unroll
            for (int t = 0; t < CHUNK_TILES; ++t) {
                v2f b = ldsB[pBase + t * 16];    // ds_load_b64, conflict-free
                acc[t] = __builtin_amdgcn_wmma_f32_16x16x4_f32(
                    false, a, false, b, (short)0, acc[t], false, false);
            }
        }

        if (active) {                            // wave-uniform: scalar branch
            #pragma unroll
            for (int t = 0; t < CHUNK_TILES; ++t) {
                const int w = colBase + t * 16 + m;
                const float bv = ADD_BIAS ? bias[w] : 0.f;
                #pragma unroll
                for (int v = 0; v < 8; ++v) {
                    const int rr = rDBase + v;
                    const int nn = rr / D;
                    const int dd = rr - nn * D;
                    out[(size_t)nn * ROW_STRIDE + blockOff + w * D + dd] =
                        acc[t][v] * scale + bv;
                }
            }
        }
    }
}

extern "C" void kernel_launch(void* const* d_in, const int* in_sizes, int n_in,
                              void* d_out, int out_size, void* d_ws, size_t ws_size,
                              hipStream_t stream) {
    (void)in_sizes; (void)n_in; (void)out_size; (void)d_ws; (void)ws_size;

    const float* x  = (const float*)d_in[0];
    const float* W0 = (const float*)d_in[1];
    const float* W1 = (const float*)d_in[2];
    const float* W2 = (const float*)d_in[3];
    const float* b  = (const float*)d_in[4];
    float* out = (float*)d_out;

    // Block 0: mul=256, l=0 (d=1), offset 0, scale 1/16, bias added.
    {
        constexpr int rowTiles = (NNODES * 1) / 16;
        dim3 grid((rowTiles + 7) / 8);
        irrep_linear_wmma<256, 1, true><<<grid, 256, 256 * 256, stream>>>(
            x, W0, b, out, 0, 0.0625f);
    }
    // Block 1: mul=128, l=1 (d=3), offset 256, scale 1/sqrt(128).
    {
        constexpr int rowTiles = (NNODES * 3) / 16;
        dim3 grid((rowTiles + 7) / 8);
        irrep_linear_wmma<128, 3, false><<<grid, 256, 128 * 256, stream>>>(
            x, W1, b, out, 256, 0.08838834764831845f);
    }
    // Block 2: mul=64, l=2 (d=5), offset 640, scale 1/8.
    {
        constexpr int rowTiles = (NNODES * 5) / 16;
        dim3 grid((rowTiles + 7) / 8);
        irrep_linear_wmma<64, 5, false><<<grid, 256, 64 * 256, stream>>>(
            x, W2, b, out, 640, 0.125f);
    }
}